// LinearAttentionTriton_52544629900005
// MI455X (gfx1250) — compile-verified
//
#include <hip/hip_runtime.h>

#define N_TOT   262144
#define DIM     64
#define TRUNK   128
#define T_TRUNKS (N_TOT / TRUNK)        // 2048

#define NBLK_A      256
#define TRUNKS_PER_A (T_TRUNKS / NBLK_A) // 8

#define SRB 76   // padded row stride (floats) for 128x64 tiles (bank-conflict-free b64 A-fetch)
#define SRS 80   // padded row stride for 64x64 S
#define SRT 20   // per-wave 16x16 score-tile stride

typedef float v2f __attribute__((ext_vector_type(2)));
typedef float v8f __attribute__((ext_vector_type(8)));

// ---------------------------------------------------------------------------
// CDNA5 async global->LDS copies via inline asm (ASYNCcnt path)
// ---------------------------------------------------------------------------
__device__ __forceinline__ unsigned lds_off_u32(const void* p) {
    // generic -> LDS address space -> 32-bit LDS byte offset
    return (unsigned)(unsigned long long)
        (__attribute__((address_space(3))) const void*)p;
}

__device__ __forceinline__ void async_b128(const float* g, float* l) {
    unsigned lo = lds_off_u32(l);
    // GV mode: per-lane 64-bit global address in VGPR pair, LDS addr in VDST
    asm volatile("global_load_async_to_lds_b128 %0, %1, off"
                 :: "v"(lo), "v"(g) : "memory");
}

#define WAIT_ASYNC(n) asm volatile("s_wait_asynccnt %0" :: "i"(n) : "memory")

// copy a row-major 128x64 f32 tile from global into padded LDS (8 b128/thread)
__device__ __forceinline__ void copy_tile128(const float* __restrict__ g,
                                             float* lds, int tid) {
#pragma unroll
    for (int c = 0; c < 8; ++c) {
        int e = (tid + c * 256) * 4;
        int r = e >> 6, col = e & 63;
        async_b128(g + e, lds + r * SRB + col);
    }
}

// copy the 64x64 S matrix into padded LDS (4 b128/thread)
__device__ __forceinline__ void copy_S(const float* __restrict__ g,
                                       float* lds, int tid) {
#pragma unroll
    for (int c = 0; c < 4; ++c) {
        int e = (tid + c * 256) * 4;
        int r = e >> 6, col = e & 63;
        async_b128(g + e, lds + r * SRS + col);
    }
}

// ---------------------------------------------------------------------------
// WMMA helpers (f32 16x16x4, wave32)
// ---------------------------------------------------------------------------
__device__ __forceinline__ v8f wmma4(v2f a, v2f b, v8f c) {
    return __builtin_amdgcn_wmma_f32_16x16x4_f32(
        false, a, false, b, (short)0, c, false, false);
}

// A operand: rows m0..m0+15 of row-major LDS buf, k-chunk base k0 (one b64)
__device__ __forceinline__ v2f fetchA(const float* buf, int sr, int m0, int k0,
                                      int l16, int kb) {
    const float* p = buf + (m0 + l16) * sr + k0 + kb;
    v2f a; a.x = p[0]; a.y = p[1];
    return a;
}

// B operand: B[k][n] = buf[(k0+..)*sr + n0 + n]
__device__ __forceinline__ v2f fetchB(const float* buf, int sr, int k0, int n0,
                                      int l16, int kb) {
    v2f b;
    b.x = buf[(k0 + kb)     * sr + n0 + l16];
    b.y = buf[(k0 + kb + 1) * sr + n0 + l16];
    return b;
}

// ---------------------------------------------------------------------------
// Kernel A: per-block partial S = sum_t V_t^T K_t, double-buffered async loads
// ---------------------------------------------------------------------------
__global__ void __launch_bounds__(256)
s_partial_kernel(const float* __restrict__ K, const float* __restrict__ V,
                 float* __restrict__ partials) {
    __shared__ __align__(16) float Kl[2][TRUNK * SRB];
    __shared__ __align__(16) float Vl[2][TRUNK * SRB];

    const int tid  = threadIdx.x;
    const int lane = tid & 31;
    const int wave = tid >> 5;
    const int l16  = lane & 15;
    const int half = lane >> 4;
    const int kb   = half * 2;

    const int mi  = wave >> 1;
    const int ni0 = (wave & 1) * 2;
    const int ni1 = ni0 + 1;

    v8f acc0 = {}; v8f acc1 = {};

    const size_t tbase = (size_t)blockIdx.x * TRUNKS_PER_A;
    // prefetch trunk 0
    copy_tile128(K + tbase * (TRUNK * DIM), Kl[0], tid);
    copy_tile128(V + tbase * (TRUNK * DIM), Vl[0], tid);

    for (int tr = 0; tr < TRUNKS_PER_A; ++tr) {
        const int cur = tr & 1;
        if (tr + 1 < TRUNKS_PER_A) {
            const size_t nb = (tbase + tr + 1) * (TRUNK * DIM);
            copy_tile128(K + nb, Kl[cur ^ 1], tid);
            copy_tile128(V + nb, Vl[cur ^ 1], tid);
            WAIT_ASYNC(16);   // 16 newer copies outstanding -> trunk tr landed
        } else {
            WAIT_ASYNC(0);
        }
        __syncthreads();      // all waves see trunk tr data

        const float* Kc = Kl[cur];
        const float* Vc = Vl[cur];
#pragma unroll
        for (int kk = 0; kk < 32; ++kk) {   // reduce over 128 trunk rows
            const int k0 = kk * 4;
            v2f a;                           // A = V^T: A[m=d][k=r] = V[r][d]
            a.x = Vc[(k0 + kb)     * SRB + mi * 16 + l16];
            a.y = Vc[(k0 + kb + 1) * SRB + mi * 16 + l16];
            v2f b0 = fetchB(Kc, SRB, k0, ni0 * 16, l16, kb);
            v2f b1 = fetchB(Kc, SRB, k0, ni1 * 16, l16, kb);
            acc0 = wmma4(a, b0, acc0);
            acc1 = wmma4(a, b1, acc1);
        }
        __syncthreads();      // done reading before buffer is overwritten
    }

    float* pp = partials + (size_t)blockIdx.x * (DIM * DIM);
#pragma unroll
    for (int g = 0; g < 8; ++g) {
        const int row = mi * 16 + half * 8 + g;
        pp[row * DIM + ni0 * 16 + l16] = acc0[g];
        pp[row * DIM + ni1 * 16 + l16] = acc1[g];
    }
}

// ---------------------------------------------------------------------------
// Kernel A2: reduce NBLK_A partials -> S (64x64, fp32)
// ---------------------------------------------------------------------------
__global__ void __launch_bounds__(256)
s_reduce_kernel(const float* __restrict__ partials, float* __restrict__ S) {
    const int e = blockIdx.x * 256 + threadIdx.x;   // 0..4095
    float s = 0.f;
    for (int p = 0; p < NBLK_A; ++p)
        s += partials[(size_t)p * (DIM * DIM) + e];
    S[e] = s;
}

// ---------------------------------------------------------------------------
// Kernel B: per trunk  out_t = (Q_t K_t^T) * (V_t S)
// ---------------------------------------------------------------------------
__global__ void __launch_bounds__(256)
intra_out_kernel(const float* __restrict__ Q, const float* __restrict__ K,
                 const float* __restrict__ V, const float* __restrict__ S,
                 float* __restrict__ out) {
    __shared__ __align__(16) float Ql[TRUNK * SRB];
    __shared__ __align__(16) float Kl[TRUNK * SRB];
    __shared__ __align__(16) float Vl[TRUNK * SRB];   // reused as W = V*S
    __shared__ __align__(16) float Sl[DIM * SRS];
    __shared__ __align__(16) float Tl[8][16 * SRT];   // per-wave score tile

    const int tid  = threadIdx.x;
    const int lane = tid & 31;
    const int wave = tid >> 5;
    const int l16  = lane & 15;
    const int half = lane >> 4;
    const int kb   = half * 2;

    const size_t base = (size_t)blockIdx.x * (TRUNK * DIM);
    copy_tile128(Q + base, Ql, tid);
    copy_tile128(K + base, Kl, tid);
    copy_tile128(V + base, Vl, tid);
    copy_S(S, Sl, tid);
    WAIT_ASYNC(0);
    __syncthreads();

    const int m0 = wave * 16;                 // this wave's 16 output rows

    // ---- stage 1: W = V @ S (rows m0..m0+15), write back over Vl ----
    v8f accW[4] = {v8f{}, v8f{}, v8f{}, v8f{}};
#pragma unroll
    for (int kk = 0; kk < 16; ++kk) {
        v2f a = fetchA(Vl, SRB, m0, kk * 4, l16, kb);
#pragma unroll
        for (int n = 0; n < 4; ++n) {
            v2f b = fetchB(Sl, SRS, kk * 4, n * 16, l16, kb);
            accW[n] = wmma4(a, b, accW[n]);
        }
    }
#pragma unroll
    for (int n = 0; n < 4; ++n)
#pragma unroll
        for (int g = 0; g < 8; ++g)
            Vl[(m0 + half * 8 + g) * SRB + n * 16 + l16] = accW[n][g];
    __syncthreads();                          // all waves need all W rows

    // ---- stage 2: out rows m0.. = sum_j scores(m0-block, j) @ W(j-block) ----
    v8f accO[4] = {v8f{}, v8f{}, v8f{}, v8f{}};
    float* tl = &Tl[wave][0];

    for (int j = 0; j < 8; ++j) {
        v8f accS = {};
#pragma unroll
        for (int kk = 0; kk < 16; ++kk) {
            v2f a = fetchA(Ql, SRB, m0,     kk * 4, l16, kb);
            v2f b = fetchA(Kl, SRB, j * 16, kk * 4, l16, kb);  // K^T == A-pattern
            accS = wmma4(a, b, accS);
        }
        // C-layout -> A-layout via per-wave padded LDS tile (wave-local; DS
        // ops from one wave are in order, compiler inserts s_wait_dscnt)
#pragma unroll
        for (int g = 0; g < 8; ++g)
            tl[(half * 8 + g) * SRT + l16] = accS[g];
#pragma unroll
        for (int kk = 0; kk < 4; ++kk) {
            v2f a = fetchA(tl, SRT, 0, kk * 4, l16, kb);
#pragma unroll
            for (int n = 0; n < 4; ++n) {
                v2f b = fetchB(Vl, SRB, j * 16 + kk * 4, n * 16, l16, kb); // W
                accO[n] = wmma4(a, b, accO[n]);
            }
        }
    }

    // ---- store output ----
    float* og = out + base;
#pragma unroll
    for (int n = 0; n < 4; ++n)
#pragma unroll
        for (int g = 0; g < 8; ++g)
            og[(m0 + half * 8 + g) * DIM + n * 16 + l16] = accO[n][g];
}

// ---------------------------------------------------------------------------
extern "C" void kernel_launch(void* const* d_in, const int* in_sizes, int n_in,
                              void* d_out, int out_size, void* d_ws, size_t ws_size,
                              hipStream_t stream) {
    const float* Q = (const float*)d_in[0];
    const float* K = (const float*)d_in[1];
    const float* V = (const float*)d_in[2];
    float* out = (float*)d_out;

    float* Sbuf     = (float*)d_ws;                 // 64*64 floats
    float* partials = Sbuf + DIM * DIM;             // NBLK_A * 64*64 floats (~4.2 MB)

    s_partial_kernel<<<NBLK_A, 256, 0, stream>>>(K, V, partials);
    s_reduce_kernel<<<(DIM * DIM) / 256, 256, 0, stream>>>(partials, Sbuf);
    intra_out_kernel<<<T_TRUNKS, 256, 0, stream>>>(Q, K, V, Sbuf, out);
}